// VMD_15985868276405
// MI455X (gfx1250) — compile-verified
//
#include <hip/hip_runtime.h>

// ---------------------------------------------------------------------------
// VMD for MI455X (gfx1250, wave32).
//  * FFT/IFFT: radix-16 Stockham, 5 stages over 2^20, each 16-pt DFT batch
//    done as fp32 complex 16x16x16 matmul via chained V_WMMA_F32_16X16X4_F32.
//  * Iteration loop: single fused elementwise + block-partial-reduce kernel
//    per iteration (L2-resident working set ~56MB < 192MB L2), plus a 1-block
//    finalize kernel for the omega update / convergence flag.
// ---------------------------------------------------------------------------

#define T_N      (1 << 20)      // 2^20 samples
#define NB_ITER  512            // blocks for the iteration kernel
#define NPART    8              // partial sums per block: num[3],den[3],dsq,psq

static constexpr float F_ALPHA = 2000.0f;
static constexpr float F_TAU   = 1e-7f;
static constexpr float F_TOL   = 1e-6f;
static constexpr float F_EPS   = 1e-8f;
static constexpr float F_PI    = 3.14159265358979323846f;
static constexpr int   MAXIT   = 50;

typedef __attribute__((ext_vector_type(2))) float v2f;
typedef __attribute__((ext_vector_type(8))) float v8f;

struct VmdCtrl { float omega[3]; int done; };

// ---------------------------------------------------------------------------
// wave32-aware block reduction (blockDim.x == 256 -> 8 waves)
// ---------------------------------------------------------------------------
__device__ __forceinline__ float blockReduceSum(float v, float* s_wave) {
  #pragma unroll
  for (int off = 16; off > 0; off >>= 1) v += __shfl_down(v, off, 32);
  const int lane = threadIdx.x & 31;
  const int wid  = threadIdx.x >> 5;
  if (lane == 0) s_wave[wid] = v;
  __syncthreads();
  float r = (threadIdx.x < (blockDim.x >> 5)) ? s_wave[threadIdx.x] : 0.0f;
  if (wid == 0) {
    #pragma unroll
    for (int off = 4; off > 0; off >>= 1) r += __shfl_down(r, off, 32);
  }
  __syncthreads();
  return r;   // valid in thread 0
}

// ---------------------------------------------------------------------------
// Init: pack x -> complex, zero lam/u, seed omega = 0.5*omega0, done = 0.
// (d_ws is poisoned once and never re-poisoned: everything read is re-init'd
//  here each call, so graph replays are deterministic.)
// ---------------------------------------------------------------------------
__global__ void vmd_init(const float* __restrict__ x,
                         const float* __restrict__ omega0,
                         float2* __restrict__ bufA,
                         float2* __restrict__ lam,
                         float2* __restrict__ u,
                         VmdCtrl* __restrict__ ctrl) {
  const int stride = gridDim.x * blockDim.x;
  for (int j = blockIdx.x * blockDim.x + threadIdx.x; j < T_N; j += stride) {
    bufA[j] = make_float2(x[j], 0.0f);
    lam[j]  = make_float2(0.0f, 0.0f);
    u[j]            = make_float2(0.0f, 0.0f);
    u[T_N + j]      = make_float2(0.0f, 0.0f);
    u[2 * T_N + j]  = make_float2(0.0f, 0.0f);
  }
  if (blockIdx.x == 0 && threadIdx.x == 0) {
    ctrl->omega[0] = 0.5f * omega0[0];
    ctrl->omega[1] = 0.5f * omega0[1];
    ctrl->omega[2] = 0.5f * omega0[2];
    ctrl->done = 0;
  }
}

// ---------------------------------------------------------------------------
// One Stockham radix-16 stage via fp32 WMMA.
//   For j in [0, N/16):  v[k] = in[j + k*N/16] * e^{sign*2pi*i*k*(j%Ns)/(16Ns)}
//   y[o] = sum_k W16[o,k] v[k],  W16[o,k] = e^{sign*2pi*i*o*k/16}
//   out[(j/Ns)*16Ns + (j%Ns) + o*Ns] = y[o]
// One wave handles a 16(o) x 16(j) tile: complex 16x16x16 matmul as 4 real
// matmuls = 16 chained V_WMMA_F32_16X16X4_F32 ops (full fp32 precision).
//
// Register layouts (per ISA 7.12.2):
//   A 16x4 f32: lanes 0-15 -> (m=lane, K=0/1 in vgpr0/1);
//               lanes 16-31 -> (m=lane-16, K=2/3).
//   B 4x16 assumed mirrored: vgpr p, lane L -> (k = 2*(L>>4)+p, n = L&15).
//   C/D 16x16: vgpr v, lanes 0-15 -> (M=v, N=lane); lanes 16-31 -> (M=v+8).
// ---------------------------------------------------------------------------
__global__ void fft_stage_wmma(const float2* __restrict__ in,
                               float2* __restrict__ out,
                               int nsLog2,          // Ns = 1<<nsLog2
                               float sign) {        // -1 fwd, +1 inverse
  const int waveId = (blockIdx.x * blockDim.x + threadIdx.x) >> 5;  // 0..4095
  const int lane   = threadIdx.x & 31;
  const int nTiles = T_N / 256;
  if (waveId >= nTiles) return;

  const int Ns   = 1 << nsLog2;
  const int hi   = lane >> 4;        // 0 or 1
  const int m    = lane & 15;        // A row / output col index
  const int n    = lane & 15;        // B col index (same lanes)
  const int j0   = waveId * 16;
  const int j    = j0 + n;
  const int jr   = j & (Ns - 1);
  const int jq   = j >> nsLog2;
  const float twScale = sign * 2.0f * F_PI / (float)(Ns * 16);

  // --- A matrix: DFT-16 (constant), real / imag / -imag, sliced K=4 ---
  v2f Ar[4], Ai[4], Ani[4];
  #pragma unroll
  for (int q = 0; q < 4; ++q) {
    const int kb = 4 * q + 2 * hi;
    #pragma unroll
    for (int p = 0; p < 2; ++p) {
      float ang = sign * 2.0f * F_PI * (float)(m * (kb + p)) * (1.0f / 16.0f);
      float s, c;
      __sincosf(ang, &s, &c);
      Ar[q][p]  = c;
      Ai[q][p]  = s;
      Ani[q][p] = -s;
    }
  }

  // --- B tile: gather 16 inputs per column with twiddle applied ---
  v2f Br[4], Bi[4];
  #pragma unroll
  for (int q = 0; q < 4; ++q) {
    #pragma unroll
    for (int p = 0; p < 2; ++p) {
      const int k = 4 * q + 2 * hi + p;
      const float2 xv = in[j + k * (T_N / 16)];
      float s, c;
      __sincosf(twScale * (float)(k * jr), &s, &c);
      Br[q][p] = xv.x * c - xv.y * s;
      Bi[q][p] = xv.x * s + xv.y * c;
    }
  }

  // --- complex 16x16x16 matmul: 4 real matmuls, each 4 chained K=4 WMMAs ---
  v8f accR = {0.f,0.f,0.f,0.f,0.f,0.f,0.f,0.f};
  v8f accI = {0.f,0.f,0.f,0.f,0.f,0.f,0.f,0.f};
  #pragma unroll
  for (int q = 0; q < 4; ++q)
    accR = __builtin_amdgcn_wmma_f32_16x16x4_f32(false, Ar[q],  false, Br[q],
                                                 (short)0, accR, false, false);
  #pragma unroll
  for (int q = 0; q < 4; ++q)
    accR = __builtin_amdgcn_wmma_f32_16x16x4_f32(false, Ani[q], false, Bi[q],
                                                 (short)0, accR, false, false);
  #pragma unroll
  for (int q = 0; q < 4; ++q)
    accI = __builtin_amdgcn_wmma_f32_16x16x4_f32(false, Ar[q],  false, Bi[q],
                                                 (short)0, accI, false, false);
  #pragma unroll
  for (int q = 0; q < 4; ++q)
    accI = __builtin_amdgcn_wmma_f32_16x16x4_f32(false, Ai[q],  false, Br[q],
                                                 (short)0, accI, false, false);

  // --- scatter results: out[(j/Ns)*16Ns + (j%Ns) + o*Ns] ---
  const int obase = jq * (Ns * 16) + jr;
  #pragma unroll
  for (int v = 0; v < 8; ++v) {
    const int o = v + 8 * hi;
    out[obase + o * Ns] = make_float2(accR[v], accI[v]);
  }
}

// ---------------------------------------------------------------------------
// Fused per-iteration kernel: u update (stale omega), power sums for omega,
// dual ascent on lam, u_diff accumulation. Frozen when ctrl->done (matches
// the jnp.where(done, old, new) gating by the *incoming* done flag).
// ---------------------------------------------------------------------------
__global__ void vmd_iter_main(const float2* __restrict__ fhat,
                              float2* __restrict__ lam,
                              float2* __restrict__ u,
                              float* __restrict__ partials,
                              const VmdCtrl* __restrict__ ctrl) {
  if (ctrl->done) return;
  __shared__ float s_wave[8];

  const float om0 = ctrl->omega[0];
  const float om1 = ctrl->omega[1];
  const float om2 = ctrl->omega[2];
  const float tau2 = F_TAU * F_TAU;
  const float invT = 1.0f / (float)T_N;

  float num0 = 0.f, num1 = 0.f, num2 = 0.f;
  float den0 = 0.f, den1 = 0.f, den2 = 0.f;
  float dsq  = 0.f, psq  = 0.f;

  const int stride = gridDim.x * blockDim.x;
  for (int j = blockIdx.x * blockDim.x + threadIdx.x; j < T_N; j += stride) {
    if (j + stride < T_N) __builtin_prefetch(&fhat[j + stride], 0, 0);

    // signed frequency of unshifted bin j == fftshifted freqs of reference
    float fr = (float)j * invT;
    if (j >= T_N / 2) fr -= 1.0f;

    const float2 fh = fhat[j];
    const float2 lm = lam[j];
    const float rr = fh.x - 0.5f * lm.x;
    const float ri = fh.y - 0.5f * lm.y;

    const float e0 = fr - om0, e1 = fr - om1, e2 = fr - om2;
    const float d0 = e0 * e0, d1 = e1 * e1, d2 = e2 * e2;
    const float q0 = 1.0f / (1.0f + F_ALPHA * (d0 + d1 + tau2));
    const float q1 = 1.0f / (1.0f + F_ALPHA * (d0 + d1 + d2 + tau2));
    const float q2 = 1.0f / (1.0f + F_ALPHA * (d1 + d2 + tau2));

    const float u0r = rr * q0, u0i = ri * q0;
    const float u1r = rr * q1, u1i = ri * q1;
    const float u2r = rr * q2, u2i = ri * q2;

    const float p0 = u0r * u0r + u0i * u0i;
    const float p1 = u1r * u1r + u1i * u1i;
    const float p2 = u2r * u2r + u2i * u2i;
    num0 += p0 * fr; den0 += p0;
    num1 += p1 * fr; den1 += p1;
    num2 += p2 * fr; den2 += p2;

    // u_diff terms against previous iterate (read before overwrite)
    const float2 pv0 = u[j];
    const float2 pv1 = u[T_N + j];
    const float2 pv2 = u[2 * T_N + j];
    float a;
    a = u0r - pv0.x; dsq += a * a; a = u0i - pv0.y; dsq += a * a;
    a = u1r - pv1.x; dsq += a * a; a = u1i - pv1.y; dsq += a * a;
    a = u2r - pv2.x; dsq += a * a; a = u2i - pv2.y; dsq += a * a;
    psq += pv0.x * pv0.x + pv0.y * pv0.y
         + pv1.x * pv1.x + pv1.y * pv1.y
         + pv2.x * pv2.x + pv2.y * pv2.y;

    u[j]           = make_float2(u0r, u0i);
    u[T_N + j]     = make_float2(u1r, u1i);
    u[2 * T_N + j] = make_float2(u2r, u2i);

    const float sr = u0r + u1r + u2r;
    const float si = u0i + u1i + u2i;
    lam[j] = make_float2(lm.x + F_TAU * (sr - fh.x),
                         lm.y + F_TAU * (si - fh.y));
  }

  float vals[NPART] = {num0, num1, num2, den0, den1, den2, dsq, psq};
  #pragma unroll
  for (int q = 0; q < NPART; ++q) {
    const float r = blockReduceSum(vals[q], s_wave);
    if (threadIdx.x == 0) partials[blockIdx.x * NPART + q] = r;
  }
}

// ---------------------------------------------------------------------------
// Finalize: omega update + convergence (every 10th iteration, sets done).
// ---------------------------------------------------------------------------
__global__ void vmd_iter_finalize(const float* __restrict__ partials,
                                  VmdCtrl* __restrict__ ctrl, int iter) {
  if (ctrl->done) return;
  __shared__ float s_wave[8];
  __shared__ float tot[NPART];

  float acc[NPART] = {0.f, 0.f, 0.f, 0.f, 0.f, 0.f, 0.f, 0.f};
  for (int b = threadIdx.x; b < NB_ITER; b += blockDim.x) {
    #pragma unroll
    for (int q = 0; q < NPART; ++q) acc[q] += partials[b * NPART + q];
  }
  #pragma unroll
  for (int q = 0; q < NPART; ++q) {
    const float r = blockReduceSum(acc[q], s_wave);
    if (threadIdx.x == 0) tot[q] = r;
  }
  __syncthreads();
  if (threadIdx.x == 0) {
    float om[3];
    #pragma unroll
    for (int k = 0; k < 3; ++k) om[k] = tot[k] / (tot[3 + k] + F_EPS);
    const float udiff = tot[6] / (tot[7] + F_EPS);
    const float odiff = (fabsf(om[0] - om[2]) + fabsf(om[1] - om[0]) +
                         fabsf(om[2] - om[1])) * (1.0f / 3.0f);
    ctrl->omega[0] = om[0];
    ctrl->omega[1] = om[1];
    ctrl->omega[2] = om[2];
    if ((iter % 10) == 0 && udiff < F_TOL && odiff < F_TOL) ctrl->done = 1;
  }
}

// ---------------------------------------------------------------------------
// Real part of IFFT result, scaled by 1/T.
// ---------------------------------------------------------------------------
__global__ void vmd_real_out(const float2* __restrict__ in,
                             float* __restrict__ out) {
  const float s = 1.0f / (float)T_N;
  const int stride = gridDim.x * blockDim.x;
  for (int j = blockIdx.x * blockDim.x + threadIdx.x; j < T_N; j += stride)
    out[j] = in[j].x * s;
}

// ---------------------------------------------------------------------------
extern "C" void kernel_launch(void* const* d_in, const int* in_sizes, int n_in,
                              void* d_out, int out_size, void* d_ws, size_t ws_size,
                              hipStream_t stream) {
  (void)in_sizes; (void)n_in; (void)out_size; (void)ws_size;

  const float* x      = (const float*)d_in[0];
  const float* omega0 = (const float*)d_in[1];
  float* out = (float*)d_out;

  // workspace layout (float2 elements): fhat | lam | u[3] | bufA | bufB | partials | ctrl
  float2* fhat = (float2*)d_ws;
  float2* lam  = fhat + T_N;
  float2* u    = lam + T_N;                  // 3 * T_N
  float2* bufA = u + 3 * T_N;
  float2* bufB = bufA + T_N;
  float*  partials = (float*)(bufB + T_N);
  VmdCtrl* ctrl = (VmdCtrl*)(partials + NB_ITER * NPART);

  const dim3 blk(256);
  const dim3 grdBig(512);

  vmd_init<<<grdBig, blk, 0, stream>>>(x, omega0, bufA, lam, u, ctrl);

  // forward FFT (sign = -1), 5 radix-16 stages; final stage lands in fhat
  fft_stage_wmma<<<grdBig, blk, 0, stream>>>(bufA, bufB,  0, -1.0f);
  fft_stage_wmma<<<grdBig, blk, 0, stream>>>(bufB, bufA,  4, -1.0f);
  fft_stage_wmma<<<grdBig, blk, 0, stream>>>(bufA, bufB,  8, -1.0f);
  fft_stage_wmma<<<grdBig, blk, 0, stream>>>(bufB, bufA, 12, -1.0f);
  fft_stage_wmma<<<grdBig, blk, 0, stream>>>(bufA, fhat, 16, -1.0f);

  // ADMM iterations (device-side done flag freezes state, matching reference)
  for (int it = 0; it < MAXIT; ++it) {
    vmd_iter_main<<<dim3(NB_ITER), blk, 0, stream>>>(fhat, lam, u, partials, ctrl);
    vmd_iter_finalize<<<dim3(1), blk, 0, stream>>>(partials, ctrl, it);
  }

  // inverse FFT (sign = +1) per mode, then real part * (1/T)
  for (int k = 0; k < 3; ++k) {
    fft_stage_wmma<<<grdBig, blk, 0, stream>>>(u + (size_t)k * T_N, bufA, 0, 1.0f);
    fft_stage_wmma<<<grdBig, blk, 0, stream>>>(bufA, bufB,  4, 1.0f);
    fft_stage_wmma<<<grdBig, blk, 0, stream>>>(bufB, bufA,  8, 1.0f);
    fft_stage_wmma<<<grdBig, blk, 0, stream>>>(bufA, bufB, 12, 1.0f);
    fft_stage_wmma<<<grdBig, blk, 0, stream>>>(bufB, bufA, 16, 1.0f);
    vmd_real_out<<<grdBig, blk, 0, stream>>>(bufA, out + (size_t)k * T_N);
  }
}